// EIGNN_w_iter_adap_gamma_52733608461012
// MI455X (gfx1250) — compile-verified
//
#include <hip/hip_runtime.h>
#include <math.h>

typedef __attribute__((ext_vector_type(2))) float v2f;
typedef __attribute__((ext_vector_type(8))) float v8f;

#define MDIM 256
#define LDS_STRIDE 260   // 256 + 4 pad: row stride 1040B -> 16B aligned, bank-conflict free

// ---------------- generic zero (grid-stride, float4) ----------------
__global__ void zero_kernel(float4* __restrict__ p, long n4) {
  long i = (long)blockIdx.x * blockDim.x + threadIdx.x;
  long stride = (long)gridDim.x * blockDim.x;
  float4 z; z.x = 0.f; z.y = 0.f; z.z = 0.f; z.w = 0.f;
  for (; i < n4; i += stride) p[i] = z;
}

// ---------------- FF = F^T F (256x256), ss += sum(FF^2) ----------------
__global__ __launch_bounds__(256) void gram_kernel(const float* __restrict__ F,
                                                   float* __restrict__ FF,
                                                   float* __restrict__ ss) {
  int i = blockIdx.x;
  int j = threadIdx.x;
  float s = 0.0f;
  for (int k = 0; k < MDIM; ++k)
    s = fmaf(F[k * MDIM + i], F[k * MDIM + j], s);
  FF[i * MDIM + j] = s;
  atomicAdd(ss, s * s);
}

// ---- Gpk[(k/2)*512 + c*2 + (k&1)] = FF[k][c] / (||FF||_F + eps) ----
__global__ __launch_bounds__(256) void pack_kernel(const float* __restrict__ FF,
                                                   const float* __restrict__ ss,
                                                   float* __restrict__ Gpk) {
  int k = blockIdx.x;
  int c = threadIdx.x;
  float scale = 1.0f / (sqrtf(ss[0]) + 1e-12f);
  Gpk[(k >> 1) * (2 * MDIM) + c * 2 + (k & 1)] = FF[k * MDIM + c] * scale;
}

// ---------------- X (M,N) -> Xt (NPAD, M) ----------------
__global__ __launch_bounds__(256) void transpose_in_kernel(const float* __restrict__ X,
                                                           float* __restrict__ Xt, int N) {
  __shared__ float t[32][33];
  int n0 = blockIdx.x * 32, m0 = blockIdx.y * 32;
  int tx = threadIdx.x, ty = threadIdx.y;
  #pragma unroll
  for (int i = 0; i < 4; ++i) {
    int m = m0 + ty + i * 8, n = n0 + tx;
    t[ty + i * 8][tx] = (n < N) ? X[(long)m * N + n] : 0.0f;
  }
  __syncthreads();
  #pragma unroll
  for (int i = 0; i < 4; ++i) {
    int n = n0 + ty + i * 8, m = m0 + tx;
    if (n < N) Xt[(long)n * MDIM + m] = t[tx][ty + i * 8];
  }
}

// ---------------- Zt (NPAD, M) -> out (M, N) ----------------
__global__ __launch_bounds__(256) void transpose_out_kernel(const float* __restrict__ Zt,
                                                            float* __restrict__ out, int N) {
  __shared__ float t[32][33];
  int n0 = blockIdx.x * 32, m0 = blockIdx.y * 32;
  int tx = threadIdx.x, ty = threadIdx.y;
  #pragma unroll
  for (int i = 0; i < 4; ++i) {
    int n = n0 + ty + i * 8, m = m0 + tx;
    t[ty + i * 8][tx] = Zt[(long)n * MDIM + m];
  }
  __syncthreads();
  #pragma unroll
  for (int i = 0; i < 4; ++i) {
    int m = m0 + ty + i * 8, n = n0 + tx;
    if (n < N) out[(long)m * N + n] = t[tx][ty + i * 8];
  }
}

// ---------------- CSR-by-destination build (once per launch) ----------------
__global__ void count_kernel(const int* __restrict__ cols, int* __restrict__ cnt, int E) {
  int e = blockIdx.x * 256 + threadIdx.x;
  if (e < E) atomicAdd(&cnt[cols[e]], 1);
}

// single-workgroup chunked exclusive scan (runs once; N=30000 -> 30 chunks)
__global__ __launch_bounds__(1024) void scan_kernel(const int* __restrict__ cnt,
                                                    int* __restrict__ off, int N) {
  __shared__ int buf[1024];
  __shared__ int carry;
  int t = threadIdx.x;
  if (t == 0) carry = 0;
  __syncthreads();
  for (int base = 0; base < N; base += 1024) {
    int i = base + t;
    int v = (i < N) ? cnt[i] : 0;
    buf[t] = v;
    __syncthreads();
    #pragma unroll
    for (int s = 1; s < 1024; s <<= 1) {
      int add = (t >= s) ? buf[t - s] : 0;
      __syncthreads();
      buf[t] += add;
      __syncthreads();
    }
    if (i < N) off[i] = carry + buf[t] - v;   // exclusive
    __syncthreads();                          // all reads of carry done
    if (t == 1023) carry += buf[1023];
    __syncthreads();
  }
  if (t == 0) off[N] = carry;
}

__global__ void fill_kernel(const int* __restrict__ cols, const int* __restrict__ off,
                            int* __restrict__ cursor, int* __restrict__ eidx, int E) {
  int e = blockIdx.x * 256 + threadIdx.x;
  if (e < E) {
    int c = cols[e];
    int p = atomicAdd(&cursor[c], 1);
    eidx[off[c] + p] = e;
  }
}

// per-bucket insertion sort -> deterministic accumulation order
__global__ void sort_kernel(const int* __restrict__ off, int* __restrict__ eidx, int N) {
  int c = blockIdx.x * 256 + threadIdx.x;
  if (c >= N) return;
  int b = off[c], e = off[c + 1];
  for (int i = b + 1; i < e; ++i) {
    int key = eidx[i];
    int j = i - 1;
    while (j >= b && eidx[j] > key) { eidx[j + 1] = eidx[j]; --j; }
    eidx[j + 1] = key;
  }
}

// ---------------- atomic-free SpMM gather: one wave32 per destination node ----------------
__global__ __launch_bounds__(256) void gather_kernel(const float* __restrict__ Zt,
                                                     const int* __restrict__ rowsArr,
                                                     const float* __restrict__ vals,
                                                     const int* __restrict__ off,
                                                     const int* __restrict__ eidx,
                                                     float* __restrict__ SZt, int N) {
  int c = (int)(((unsigned)blockIdx.x * 256u + threadIdx.x) >> 5);
  int lane = threadIdx.x & 31;
  if (c >= N) return;
  int jb = off[c], je = off[c + 1];
  float4 a0; a0.x = a0.y = a0.z = a0.w = 0.f;
  float4 a1 = a0;
  for (int j = jb; j < je; ++j) {
    int e = eidx[j];
    int r = rowsArr[e];
    float v = vals[e];
    const float4* __restrict__ src = (const float4*)(Zt + (long)r * MDIM);
    float4 z0 = src[lane];
    float4 z1 = src[lane + 32];
    a0.x = fmaf(v, z0.x, a0.x); a0.y = fmaf(v, z0.y, a0.y);
    a0.z = fmaf(v, z0.z, a0.z); a0.w = fmaf(v, z0.w, a0.w);
    a1.x = fmaf(v, z1.x, a1.x); a1.y = fmaf(v, z1.y, a1.y);
    a1.z = fmaf(v, z1.z, a1.z); a1.w = fmaf(v, z1.w, a1.w);
  }
  float4* __restrict__ dst = (float4*)(SZt + (long)c * MDIM);
  dst[lane] = a0;
  dst[lane + 32] = a1;
}

// ---------------- Zt = gamma * SZt @ G + Xt  (WMMA f32 16x16x4) ----------------
__global__ __launch_bounds__(256) void gemm_kernel(const float* __restrict__ SZt,
                                                   const float* __restrict__ Gpk,
                                                   const float* __restrict__ Xt,
                                                   const float* __restrict__ gamma,
                                                   float* __restrict__ Zt) {
  __shared__ float tileA[32 * LDS_STRIDE];   // single shared var -> LDS offset 0
  const int tid = threadIdx.x;
  const long rowBase = (long)blockIdx.x * 32;

  // Async-stage 32x256 slab of SZt straight into LDS (no VGPR round trip).
  // GVS mode: saddr(SGPR pair) + 32-bit per-lane offset; LDS dest address in VGPR.
  // The LDS address is taken as ptrtoint of the real element pointer: low 32 bits of a
  // flat LDS pointer are the wave-relative LDS byte offset (aperture is in the high
  // half), and the ptrtoint captures tileA so the "memory" clobber covers it — this is
  // what keeps the later ds_loads from being folded to undef.
  {
    unsigned long long srcBase = (unsigned long long)(const void*)(SZt + rowBase * MDIM);
    #pragma unroll
    for (int it = 0; it < 8; ++it) {
      int idx = tid + it * 256;                 // float4 index in [0, 2048)
      int r = idx >> 6;
      int c4 = idx & 63;
      unsigned ldsAddr =
          (unsigned)(unsigned long long)(const void*)&tileA[r * LDS_STRIDE + c4 * 4];
      unsigned gOff = (unsigned)(idx * 16);
      asm volatile("global_load_async_to_lds_b128 %0, %1, %2"
                   :: "v"(ldsAddr), "v"(gOff), "s"(srcBase)
                   : "memory");
    }
    asm volatile("s_wait_asynccnt 0x0" :: "v"((const void*)tileA) : "memory");
  }
  __syncthreads();

  const int lane = tid & 31;
  const int wid  = tid >> 5;
  const int row0 = (wid & 1) * 16;
  const int col0 = (wid >> 1) * 64;
  const int m    = lane & 15;
  const int hi   = lane >> 4;
  const int kb   = hi * 2;

  v8f acc0 = {}, acc1 = {}, acc2 = {}, acc3 = {};
  const v2f* __restrict__ Gp2 = (const v2f*)Gpk;
  const float* aRow = &tileA[(row0 + m) * LDS_STRIDE + kb];
  const int cb = col0 + m;

  #pragma unroll 4
  for (int k = 0; k < MDIM; k += 4) {
    v2f a = *(const v2f*)&aRow[k];
    int kp = (k >> 1) + hi;
    v2f b0 = Gp2[kp * MDIM + cb +  0];
    v2f b1 = Gp2[kp * MDIM + cb + 16];
    v2f b2 = Gp2[kp * MDIM + cb + 32];
    v2f b3 = Gp2[kp * MDIM + cb + 48];
    acc0 = __builtin_amdgcn_wmma_f32_16x16x4_f32(false, a, false, b0, (short)0, acc0, false, false);
    acc1 = __builtin_amdgcn_wmma_f32_16x16x4_f32(false, a, false, b1, (short)0, acc1, false, false);
    acc2 = __builtin_amdgcn_wmma_f32_16x16x4_f32(false, a, false, b2, (short)0, acc2, false, false);
    acc3 = __builtin_amdgcn_wmma_f32_16x16x4_f32(false, a, false, b3, (short)0, acc3, false, false);
  }

  float g = fminf(fmaxf(gamma[0], 0.0f), 1.0f);
  long grow = rowBase + row0 + hi * 8;
  #pragma unroll
  for (int i = 0; i < 8; ++i) {
    long base = (grow + i) * MDIM + cb;
    Zt[base +  0] = fmaf(g, acc0[i], Xt[base +  0]);
    Zt[base + 16] = fmaf(g, acc1[i], Xt[base + 16]);
    Zt[base + 32] = fmaf(g, acc2[i], Xt[base + 32]);
    Zt[base + 48] = fmaf(g, acc3[i], Xt[base + 48]);
  }
}

extern "C" void kernel_launch(void* const* d_in, const int* in_sizes, int n_in,
                              void* d_out, int out_size, void* d_ws, size_t ws_size,
                              hipStream_t stream) {
  const float* F     = (const float*)d_in[0];
  const float* gamma = (const float*)d_in[1];
  const float* X     = (const float*)d_in[2];
  const float* vals  = (const float*)d_in[3];
  const int*   rows  = (const int*)d_in[4];
  const int*   cols  = (const int*)d_in[5];
  float* out = (float*)d_out;

  const int N = in_sizes[2] / MDIM;     // 30000
  const int E = in_sizes[4];            // 480000
  const int nTiles = (N + 31) / 32;     // 938
  const long NPAD = (long)nTiles * 32;  // 30016
  const long featBytes = NPAD * MDIM * sizeof(float);

  char* ws = (char*)d_ws;
  float* Xt  = (float*)(ws);
  float* Zt  = (float*)(ws + featBytes);
  float* SZt = (float*)(ws + 2 * featBytes);
  float* FF  = (float*)(ws + 3 * featBytes);
  float* Gpk = FF + MDIM * MDIM;
  float* ss  = Gpk + MDIM * MDIM;             // 256B region
  int*   off    = (int*)(ws + 3 * featBytes + 2 * MDIM * MDIM * sizeof(float) + 256);
  int*   cursor = off + (N + 64);             // padded gaps absorb float4-granular zeroing
  int*   eidx   = cursor + (N + 64);

  const long feat4 = NPAD * MDIM / 4;
  const int zGrid = 2048;
  const int eGrid = (E + 255) / 256;
  const int nGrid = (N + 255) / 256;

  // G = F^T F / (||F^T F||_F + eps), packed for B-fragments
  zero_kernel<<<1, 32, 0, stream>>>((float4*)ss, 1);
  gram_kernel<<<MDIM, MDIM, 0, stream>>>(F, FF, ss);
  pack_kernel<<<MDIM, MDIM, 0, stream>>>(FF, ss, Gpk);

  // Xt = X^T (padded rows zeroed), Zt = 0, SZt = 0 (padded tail stays zero forever)
  zero_kernel<<<zGrid, 256, 0, stream>>>((float4*)Xt, feat4);
  transpose_in_kernel<<<dim3(nTiles, MDIM / 32), dim3(32, 8), 0, stream>>>(X, Xt, N);
  zero_kernel<<<zGrid, 256, 0, stream>>>((float4*)Zt, feat4);
  zero_kernel<<<zGrid, 256, 0, stream>>>((float4*)SZt, feat4);

  // CSR-by-destination: count -> scan -> fill -> per-bucket sort (deterministic order)
  zero_kernel<<<64, 256, 0, stream>>>((float4*)cursor, (N + 3) / 4);
  count_kernel<<<eGrid, 256, 0, stream>>>(cols, cursor, E);
  scan_kernel<<<1, 1024, 0, stream>>>(cursor, off, N);
  zero_kernel<<<64, 256, 0, stream>>>((float4*)cursor, (N + 3) / 4);
  fill_kernel<<<eGrid, 256, 0, stream>>>(cols, off, cursor, eidx, E);
  sort_kernel<<<nGrid, 256, 0, stream>>>(off, eidx, N);

  // 30 fixed-point iterations + 1 extra training-mode step, all L2-resident, no atomics
  const int gatherGrid = (N + 7) / 8;   // 8 waves (dest nodes) per block
  for (int it = 0; it < 31; ++it) {
    gather_kernel<<<gatherGrid, 256, 0, stream>>>(Zt, rows, vals, off, eidx, SZt, N);
    gemm_kernel<<<nTiles, 256, 0, stream>>>(SZt, Gpk, Xt, gamma, Zt);
  }

  // out (M,N) = Zt^T
  transpose_out_kernel<<<dim3(nTiles, MDIM / 32), dim3(32, 8), 0, stream>>>(Zt, out, N);
}